// HeadAttentionMultiLatent_12601434046471
// MI455X (gfx1250) — compile-verified
//
#include <hip/hip_runtime.h>

typedef float v2f __attribute__((ext_vector_type(2)));
typedef float v8f __attribute__((ext_vector_type(8)));

#define B_    32
#define SPREV 4096
#define LSEQ  4097
#define DM    512
#define NH    8
#define HD    64
#define CHUNK 513            // 8 chunks cover 4097
#define NCHUNK 8

// ---------------------------------------------------------------------------
// GEMM: Out[32,512] = A[32,512] @ W[512,512] (+ bias), fp32 WMMA 16x16x4.
// 64 wave-tiles (2 Mtiles x 32 Ntiles), 4 waves / block, 16 blocks.
// ---------------------------------------------------------------------------
__global__ __launch_bounds__(128) void gemm_32x512x512(
    const float* __restrict__ A, const float* __restrict__ W,
    const float* __restrict__ bias, float* __restrict__ Out, int has_bias) {
  int wave   = blockIdx.x * 4 + (threadIdx.x >> 5);
  int tile_m = wave >> 5;
  int tile_n = wave & 31;
  int lane   = threadIdx.x & 31;
  int half   = lane >> 4;
  int lm     = lane & 15;

  const int m0 = tile_m * 16;
  const int n0 = tile_n * 16;
  v8f acc = {};
#pragma unroll 4
  for (int kb = 0; kb < 128; ++kb) {
    int k = kb * 4 + half * 2;
    v2f a, b;
    a.x = A[(m0 + lm) * DM + k];
    a.y = A[(m0 + lm) * DM + k + 1];
    b.x = W[(size_t)k * DM + n0 + lm];
    b.y = W[(size_t)(k + 1) * DM + n0 + lm];
    acc = __builtin_amdgcn_wmma_f32_16x16x4_f32(
        false, a, false, b, (short)0, acc, false, false);
  }
  int n = n0 + lm;
  float bv = has_bias ? bias[n] : 0.0f;
#pragma unroll
  for (int r = 0; r < 8; ++r) {
    int m = m0 + half * 8 + r;
    Out[m * DM + n] = acc[r] + bv;
  }
}

// ---------------------------------------------------------------------------
// Fused single-pass kernel: per (chunk, h, b) block, stream the head-slice of
// K and V exactly once from HBM; emit k_combined/v_combined and split-head
// k/v output segments; compute logits (shfl-xor wave dot) -> qk output; do
// online (flash) softmax accumulation of z. Partials (m, l, z[64]) -> ws.
// grid=(8,8,32), block=256 (8 wave32s).
// ---------------------------------------------------------------------------
__global__ __launch_bounds__(256) void fused_kv_attn(
    const float* __restrict__ kprev, const float* __restrict__ vprev,
    const float* __restrict__ knew,  const float* __restrict__ vnew,
    const float* __restrict__ qv,    const float* __restrict__ mask,
    float* __restrict__ kc, float* __restrict__ vc,
    float* __restrict__ ks, float* __restrict__ vs,
    float* __restrict__ qk_out, float* __restrict__ part) {
  __shared__ float qsh[64];
  __shared__ float wm[8], wl[8];
  __shared__ float zp[8][64];

  int c = blockIdx.x;          // s-chunk
  int h = blockIdx.y;
  int b = blockIdx.z;
  int tid = threadIdx.x;
  int w = tid >> 5, lane = tid & 31;

  if (tid < 64) qsh[tid] = qv[((size_t)b * NH + h) * HD + tid];
  __syncthreads();
  float q0 = qsh[2 * lane], q1 = qsh[2 * lane + 1];

  const float scale = 0.125f;  // 1/sqrt(64)
  int s_begin = c * CHUNK;
  int s_end   = (s_begin + CHUNK < LSEQ) ? s_begin + CHUNK : LSEQ;

  float m = -3.0e38f, l = 0.0f;
  v2f acc; acc.x = 0.0f; acc.y = 0.0f;

  for (int s = s_begin + w; s < s_end; s += 8) {
    const float* ksrc;
    const float* vsrc;
    if (s < SPREV) {
      ksrc = kprev + ((size_t)b * SPREV + s) * DM + h * HD;
      vsrc = vprev + ((size_t)b * SPREV + s) * DM + h * HD;
    } else {
      ksrc = knew + (size_t)b * DM + h * HD;
      vsrc = vnew + (size_t)b * DM + h * HD;
    }
    v2f kvv = *reinterpret_cast<const v2f*>(ksrc + 2 * lane);
    v2f vvv = *reinterpret_cast<const v2f*>(vsrc + 2 * lane);

    // materialize combined + split-head layouts (each segment written once)
    *reinterpret_cast<v2f*>(kc + ((size_t)b * LSEQ + s) * DM + h * HD + 2 * lane) = kvv;
    *reinterpret_cast<v2f*>(vc + ((size_t)b * LSEQ + s) * DM + h * HD + 2 * lane) = vvv;
    *reinterpret_cast<v2f*>(ks + (((size_t)b * NH + h) * LSEQ + s) * (size_t)HD + 2 * lane) = kvv;
    *reinterpret_cast<v2f*>(vs + (((size_t)b * NH + h) * LSEQ + s) * (size_t)HD + 2 * lane) = vvv;

    // logit = scale * q . k  (wave butterfly reduction -> uniform)
    float p = kvv.x * q0 + kvv.y * q1;
    p += __shfl_xor(p, 16);
    p += __shfl_xor(p, 8);
    p += __shfl_xor(p, 4);
    p += __shfl_xor(p, 2);
    p += __shfl_xor(p, 1);
    float logit = p * scale;
    if (lane == 0) qk_out[((size_t)b * NH + h) * LSEQ + s] = logit;

    float ml = logit + fminf(1.0f, mask[(size_t)b * LSEQ + s]) * (-1.0e9f);

    // online softmax update (wave-uniform m,l; lane-held z dims)
    float mnew = fmaxf(m, ml);
    float corr = __expf(m - mnew);
    float pw   = __expf(ml - mnew);
    l = l * corr + pw;
    acc.x = acc.x * corr + pw * vvv.x;
    acc.y = acc.y * corr + pw * vvv.y;
    m = mnew;
  }

  if (lane == 0) { wm[w] = m; wl[w] = l; }
  zp[w][2 * lane]     = acc.x;
  zp[w][2 * lane + 1] = acc.y;
  __syncthreads();

  // combine the 8 waves -> one partial record per (b,h,chunk)
  if (tid < 64) {
    float mb = wm[0];
    for (int i = 1; i < 8; ++i) mb = fmaxf(mb, wm[i]);
    float zsum = 0.0f, lsum = 0.0f;
    for (int i = 0; i < 8; ++i) {
      float f = __expf(wm[i] - mb);
      zsum += f * zp[i][tid];
      lsum += f * wl[i];
    }
    float* rec = part + (((size_t)b * NH + h) * NCHUNK + c) * 66;
    if (tid == 0) { rec[0] = mb; rec[1] = lsum; }
    rec[2 + tid] = zsum;
  }
}

// ---------------------------------------------------------------------------
// Finalize: combine chunk partials -> z; renormalize logits -> attn weights.
// grid=(8,32), block=256.
// ---------------------------------------------------------------------------
__global__ __launch_bounds__(256) void attn_finalize(
    const float* __restrict__ part, const float* __restrict__ qk_out,
    const float* __restrict__ mask, float* __restrict__ attn_out,
    float* __restrict__ z_ws) {
  __shared__ float gms[2];
  int h = blockIdx.x, b = blockIdx.y;
  int tid = threadIdx.x;
  const float* base = part + ((size_t)b * NH + h) * NCHUNK * 66;

  if (tid < 64) {
    float gmax = base[0];
    for (int c = 1; c < NCHUNK; ++c) gmax = fmaxf(gmax, base[c * 66]);
    float gsum = 0.0f, z = 0.0f;
    for (int c = 0; c < NCHUNK; ++c) {
      float f = __expf(base[c * 66] - gmax);
      gsum += f * base[c * 66 + 1];
      z    += f * base[c * 66 + 2 + tid];
    }
    z_ws[(size_t)b * DM + (size_t)h * HD + tid] = z / gsum;
    if (tid == 0) { gms[0] = gmax; gms[1] = gsum; }
  }
  __syncthreads();
  float gmax = gms[0];
  float inv  = 1.0f / gms[1];
  for (int i = tid; i < LSEQ; i += 256) {
    float logit = qk_out[((size_t)b * NH + h) * LSEQ + i];
    float ml = logit + fminf(1.0f, mask[(size_t)b * LSEQ + i]) * (-1.0e9f);
    attn_out[((size_t)b * NH + h) * LSEQ + i] = __expf(ml - gmax) * inv;
  }
}

// ---------------------------------------------------------------------------
extern "C" void kernel_launch(void* const* d_in, const int* in_sizes, int n_in,
                              void* d_out, int out_size, void* d_ws, size_t ws_size,
                              hipStream_t stream) {
  const float* x       = (const float*)d_in[0];
  const float* k_prev  = (const float*)d_in[1];
  const float* v_prev  = (const float*)d_in[2];
  const float* mask    = (const float*)d_in[3];
  const float* wq_w    = (const float*)d_in[4];
  const float* wq_b    = (const float*)d_in[5];
  const float* we_w    = (const float*)d_in[6];
  const float* we_b    = (const float*)d_in[7];
  const float* wk_w    = (const float*)d_in[8];
  const float* wv_w    = (const float*)d_in[9];
  const float* dense_w = (const float*)d_in[10];
  const float* dense_b = (const float*)d_in[11];
  float* out = (float*)d_out;
  float* ws  = (float*)d_ws;

  // output tuple offsets (flattened, in return order)
  const size_t off_attn = (size_t)B_ * DM;
  const size_t off_qk   = off_attn + (size_t)B_ * NH * LSEQ;
  const size_t off_q    = off_qk + (size_t)B_ * NH * LSEQ;
  const size_t off_k    = off_q + (size_t)B_ * NH * HD;   // q == q_latent layout
  const size_t off_v    = off_k + (size_t)B_ * NH * LSEQ * HD;
  const size_t off_kc   = off_v + (size_t)B_ * NH * LSEQ * HD;
  const size_t off_vc   = off_kc + (size_t)B_ * LSEQ * DM;

  // workspace
  float* c_new = ws;                 // 32x512
  float* k_new = ws + 16384;         // 32x512
  float* v_new = ws + 32768;         // 32x512
  float* z     = ws + 49152;         // 32x512
  float* part  = ws + 65536;         // 256 * 8 * 66 floats

  // projections (WMMA)
  gemm_32x512x512<<<16, 128, 0, stream>>>(x, wq_w, wq_b, out + off_q, 1);
  gemm_32x512x512<<<16, 128, 0, stream>>>(x, we_w, we_b, c_new, 1);
  gemm_32x512x512<<<16, 128, 0, stream>>>(c_new, wk_w, nullptr, k_new, 0);
  gemm_32x512x512<<<16, 128, 0, stream>>>(c_new, wv_w, nullptr, v_new, 0);

  // single-pass KV materialization + flash attention (bandwidth-dominant)
  fused_kv_attn<<<dim3(NCHUNK, NH, B_), 256, 0, stream>>>(
      k_prev, v_prev, k_new, v_new, out + off_q, mask,
      out + off_kc, out + off_vc, out + off_k, out + off_v,
      out + off_qk, part);

  // combine partials, emit z and attention weights
  attn_finalize<<<dim3(NH, B_), 256, 0, stream>>>(
      part, out + off_qk, mask, out + off_attn, z);

  // final dense (WMMA)
  gemm_32x512x512<<<16, 128, 0, stream>>>(z, dense_w, dense_b, out, 1);
}